// GatedGCN_21887153340606
// MI455X (gfx1250) — compile-verified
//
#include <hip/hip_runtime.h>
#include <hip/hip_bf16.h>
#include <math.h>

typedef __attribute__((ext_vector_type(2))) float v2f;
typedef __attribute__((ext_vector_type(8))) float v8f;

#define GCN_N 50000
#define GCN_E 800000
#define GCN_K 256

// ---------------------------------------------------------------------------
// Dual GEMM: out_w = A @ W, out_g = A @ G   (A: [M,K] fp32 row-major,
// W/G: [K,DOUT] row-major). One block = MT*16 rows of A staged in LDS,
// 8 waves each own TPW 16-col tiles across both outputs; every B fragment
// is reused by MT WMMAs (doubles matrix-op density vs 16-row blocks).
// fp32 WMMA 16x16x4 keeps full fp32 precision.
// ---------------------------------------------------------------------------
template <int K, int DOUT, int TPW, int MT>
__global__ __launch_bounds__(256) void gemm_dual_wmma(
    const float* __restrict__ A,
    const float* __restrict__ W,
    const float* __restrict__ G,
    float* __restrict__ outW,
    float* __restrict__ outG,
    int M)
{
    constexpr int ROWS = 16 * MT;
    constexpr int LDSS = K + 4;      // stride 260 % 64 banks = 4 -> conflict-free column reads
    __shared__ float tile[ROWS * LDSS];

    const int m0  = blockIdx.x * ROWS;
    const int tid = threadIdx.x;

    // Stage A tile (ROWS x K). blockDim.x == K == 256: thread t loads column t
    // of each row; clamp row index for the ragged final block.
    #pragma unroll
    for (int i = 0; i < ROWS; ++i) {
        int r = m0 + i;
        if (r > M - 1) r = M - 1;
        tile[i * LDSS + tid] = A[(size_t)r * K + tid];
    }
    __syncthreads();

    const int wave = tid >> 5;
    const int lane = tid & 31;
    const int half = lane >> 4;      // 0: lanes 0-15, 1: lanes 16-31
    const int l16  = lane & 15;

    v8f acc[MT][TPW];
    #pragma unroll
    for (int mt = 0; mt < MT; ++mt)
        #pragma unroll
        for (int t = 0; t < TPW; ++t) acc[mt][t] = {};

    // A fragment base: lane<16 reads K=k,k+1 of its row; lane>=16 reads K=k+2,k+3
    const float* lrow = &tile[l16 * LDSS + 2 * half];

    for (int k = 0; k < K; k += 4) {
        v2f a[MT];
        #pragma unroll
        for (int mt = 0; mt < MT; ++mt) {
            a[mt].x = lrow[mt * 16 * LDSS + k];
            a[mt].y = lrow[mt * 16 * LDSS + k + 1];
        }
        #pragma unroll
        for (int t = 0; t < TPW; ++t) {
            const int tileIdx = wave * TPW + t;
            const bool isW = (tileIdx < DOUT / 16);
            const float* Bmat = isW ? W : G;
            const int n0 = (isW ? tileIdx : tileIdx - DOUT / 16) * 16;
            // B fragment: lane<16 -> rows k,k+1 col n0+l16 ; lane>=16 -> rows k+2,k+3
            const float* bp = Bmat + (size_t)(k + 2 * half) * DOUT + n0 + l16;
            v2f b;
            b.x = bp[0];
            b.y = bp[DOUT];
            #pragma unroll
            for (int mt = 0; mt < MT; ++mt)
                acc[mt][t] = __builtin_amdgcn_wmma_f32_16x16x4_f32(
                    false, a[mt], false, b, (short)0, acc[mt][t], false, false);
        }
    }

    // C layout: VGPR r -> M = m0 + 16*mt + 8*half + r, N = n0 + l16
    #pragma unroll
    for (int mt = 0; mt < MT; ++mt) {
        #pragma unroll
        for (int t = 0; t < TPW; ++t) {
            const int tileIdx = wave * TPW + t;
            const bool isW = (tileIdx < DOUT / 16);
            float* O = isW ? outW : outG;
            const int n0 = (isW ? tileIdx : tileIdx - DOUT / 16) * 16;
            const int mbase = m0 + 16 * mt + 8 * half;
            float* op = O + (size_t)mbase * DOUT + n0 + l16;
            #pragma unroll
            for (int r = 0; r < 8; ++r)
                if (mbase + r < M)
                    op[(size_t)r * DOUT] = acc[mt][t][r];
        }
    }
}

// ---------------------------------------------------------------------------
// COO SpMM scatter: one wave per edge, lane-strided over D columns,
// atomic f32 add into L2-resident output.
// ---------------------------------------------------------------------------
template <int D>
__global__ __launch_bounds__(256) void spmm_scatter(
    const int* __restrict__ rows,
    const int* __restrict__ cols,
    const float* __restrict__ vals,
    const float* __restrict__ dense,
    float* __restrict__ out,
    int numE)
{
    const int warp = (int)((blockIdx.x * blockDim.x + threadIdx.x) >> 5);
    const int lane = threadIdx.x & 31;
    if (warp >= numE) return;
    const int   r = rows[warp];
    const int   c = cols[warp];
    const float v = vals[warp];
    const float* src = dense + (size_t)c * D;
    float*       dst = out   + (size_t)r * D;
    #pragma unroll
    for (int d = lane; d < D; d += 32)
        atomicAdd(&dst[d], v * src[d]);
}

// ---------------------------------------------------------------------------
// out = sigmoid(gate) * sup, optional ReLU
// ---------------------------------------------------------------------------
__global__ __launch_bounds__(256) void gated_act(
    const float* __restrict__ sup,
    const float* __restrict__ gate,
    float* __restrict__ out,
    size_t n, int do_relu)
{
    size_t i = (size_t)blockIdx.x * blockDim.x + threadIdx.x;
    const size_t stride = (size_t)gridDim.x * blockDim.x;
    for (; i < n; i += stride) {
        const float g = gate[i];
        const float s = sup[i];
        float y = s * (1.0f / (1.0f + __expf(-g)));
        if (do_relu) y = fmaxf(y, 0.0f);
        out[i] = y;
    }
}

// ---------------------------------------------------------------------------
// Orchestration
// ---------------------------------------------------------------------------
extern "C" void kernel_launch(void* const* d_in, const int* in_sizes, int n_in,
                              void* d_out, int out_size, void* d_ws, size_t ws_size,
                              hipStream_t stream) {
    (void)in_sizes; (void)n_in; (void)out_size; (void)ws_size;

    const float* x    = (const float*)d_in[0];   // [N,256]
    const int*   rows = (const int*)  d_in[1];   // [E]
    const int*   cols = (const int*)  d_in[2];   // [E]
    const float* vals = (const float*)d_in[3];   // [E]
    const float* W1   = (const float*)d_in[4];   // [256,256]
    const float* G1   = (const float*)d_in[5];   // [256,256]
    const float* W2   = (const float*)d_in[6];   // [256,128]
    const float* G2   = (const float*)d_in[7];   // [256,128]
    float* out = (float*)d_out;                  // [N,128]

    const size_t NB = (size_t)GCN_N * 256;       // elements per big buffer
    float* buf0 = (float*)d_ws;                  // sup1  -> later h
    float* buf1 = buf0 + NB;                     // gate1 -> later sup2|gate2
    float* buf2 = buf1 + NB;                     // agg_sup1 -> later agg2 pair
    float* buf3 = buf2 + NB;                     // agg_gate1

    const int mBlocks = (GCN_N + 31) / 32;       // 1563 (32 rows per block)
    const int spmmBlocks = (GCN_E * 32 + 255) / 256;

    // ---- Layer 1: sup1 = x@W1, gate1 = x@G1 (fused dual GEMM) ----
    gemm_dual_wmma<256, 256, 4, 2><<<mBlocks, 256, 0, stream>>>(x, W1, G1, buf0, buf1, GCN_N);

    // ---- SpMM aggregate both ----
    hipMemsetAsync(buf2, 0, NB * sizeof(float), stream);
    hipMemsetAsync(buf3, 0, NB * sizeof(float), stream);
    spmm_scatter<256><<<spmmBlocks, 256, 0, stream>>>(rows, cols, vals, buf0, buf2, GCN_E);
    spmm_scatter<256><<<spmmBlocks, 256, 0, stream>>>(rows, cols, vals, buf1, buf3, GCN_E);

    // ---- h = relu(sigmoid(agg_gate1) * agg_sup1) ----
    gated_act<<<4096, 256, 0, stream>>>(buf2, buf3, buf0, NB, 1);

    // ---- Layer 2: sup2 = h@W2, gate2 = h@G2 (both live inside buf1) ----
    const size_t NB2 = (size_t)GCN_N * 128;
    float* sup2 = buf1;
    float* gat2 = buf1 + NB2;
    gemm_dual_wmma<256, 128, 2, 2><<<mBlocks, 256, 0, stream>>>(buf0, W2, G2, sup2, gat2, GCN_N);

    // ---- SpMM aggregate both (into buf2 halves) ----
    float* agg_s2 = buf2;
    float* agg_g2 = buf2 + NB2;
    hipMemsetAsync(buf2, 0, NB * sizeof(float), stream);
    spmm_scatter<128><<<spmmBlocks, 256, 0, stream>>>(rows, cols, vals, sup2, agg_s2, GCN_E);
    spmm_scatter<128><<<spmmBlocks, 256, 0, stream>>>(rows, cols, vals, gat2, agg_g2, GCN_E);

    // ---- logits = sigmoid(agg_gate2) * agg_sup2 ----
    gated_act<<<4096, 256, 0, stream>>>(agg_s2, agg_g2, out, NB2, 0);
}